// DeChunkLayer_3367254360086
// MI455X (gfx1250) — compile-verified
//
#include <hip/hip_runtime.h>
#include <hip/hip_bf16.h>
#include <stdint.h>

// DeChunkLayer on MI455X (gfx1250).
//
// Math shortcut (exact in real arithmetic, ~ulp-level vs the reference's
// tree-order associative scan): within each chunk the EMA's fixed point is the
// chunk state itself, and every boundary resets y to that state, so
//     y[b,s,:] = (cum[b,s] > 0) ? compressed[b, cum[b,s]-1, :] : 0
// where cum = inclusive cumsum of boundary_mask along s. boundary_prob cancels.
//
// Pure data movement (~256 MiB HBM traffic -> ~11 us at 23.3 TB/s). The row
// gather uses the CDNA5 async global<->LDS engine (ASYNCcnt-tracked):
// global_load_async_to_lds_b128 + global_store_async_from_lds_b128; data never
// touches VGPRs, 32 lanes x 16 B = 512 B per wave32 instruction. Output stores
// are non-temporal so the gather source (128 MiB, ~2x reuse) stays in the
// 192 MB L2 while the write-once output streams past it.

#define B_ 8
#define S_ 4096
#define D_ 1024
#define ROW_BYTES (D_ * 4)          // 4096 bytes per row
#define WAVES_PER_BLOCK 8
#define LDS_BYTES (WAVES_PER_BLOCK * ROW_BYTES)  // 32 KB dynamic LDS
#define COPY_ITERS (ROW_BYTES / (32 * 16))       // 8 x (32 lanes * 16B)

typedef float v4f __attribute__((ext_vector_type(4)));

// ---------------------------------------------------------------------------
// Row gather via async global->LDS->global copies. One wave32 per output row
// (4 KB). Each wave owns a private 4 KB slice of the 32 KB *dynamic* LDS
// allocation (no static __shared__ vars -> dynamic region starts at LDS byte
// offset 0, so raw integer LDS addresses feed the async instructions:
// LDS address = VGPR value plus instruction offset).  Defined FIRST so the disasm snippet
// shows the async-tensor instructions.
// ---------------------------------------------------------------------------
__global__ void __launch_bounds__(32 * WAVES_PER_BLOCK)
dechunk_gather_kernel(const float* __restrict__ comp,
                      const int* __restrict__ idx,
                      float* __restrict__ y) {
    const int wave = threadIdx.x >> 5;      // wave32 on gfx1250
    const int lane = threadIdx.x & 31;

    const long long row = (long long)blockIdx.x * WAVES_PER_BLOCK + wave; // [0, B*S)
    const int b = (int)(row >> 12);         // row / 4096
    const int k = idx[row];                 // uniform across the wave

    float* dst = y + row * (long long)D_;
    unsigned long long dstp = (unsigned long long)(uintptr_t)dst + (unsigned)(lane * 16);

    if (k < 0) {
        // Before the first boundary: y = 0. Nontemporal B128 zero-fill.
        v4f z = (v4f)0.0f;
        v4f* d4 = (v4f*)dst + lane;
#pragma unroll
        for (int i = 0; i < COPY_ITERS; ++i)
            __builtin_nontemporal_store(z, d4 + i * 32);
        return;                              // wave-uniform branch
    }

    const float* src = comp + ((long long)b * S_ + k) * (long long)D_;
    unsigned long long srcp = (unsigned long long)(uintptr_t)src + (unsigned)(lane * 16);
    const unsigned lbase = (unsigned)(wave * ROW_BYTES + lane * 16);

    // Stage the 4 KB row into this wave's LDS slice with the async engine.
    // Default RT temporal hint: chunk rows are re-read (~2x) and should sit
    // in L2.
#pragma unroll
    for (int i = 0; i < COPY_ITERS; ++i) {
        asm volatile("global_load_async_to_lds_b128 %0, %1, off"
                     :
                     : "v"(lbase + (unsigned)(i * 512)),
                       "v"(srcp + (unsigned long long)(i * 512))
                     : "memory");
    }
    // Async loads decrement ASYNCcnt when the LDS write has landed.
    asm volatile("s_wait_asynccnt 0x0" ::: "memory");

    // Stream the row out to the contiguous output location; non-temporal so
    // the write-once output doesn't evict the gather source from L2.
#pragma unroll
    for (int i = 0; i < COPY_ITERS; ++i) {
        asm volatile("global_store_async_from_lds_b128 %0, %1, off th:TH_STORE_NT"
                     :
                     : "v"(dstp + (unsigned long long)(i * 512)),
                       "v"(lbase + (unsigned)(i * 512))
                     : "memory");
    }
    // Drain before the wave retires (S_ENDPGM also implies wait-idle).
    asm volatile("s_wait_asynccnt 0x0" ::: "memory");
}

// ---------------------------------------------------------------------------
// Per-batch inclusive cumsum of the boundary mask.
// Writes idx[b*S+s] = cum-1  (== -1 when no boundary seen yet).
// One 256-thread workgroup per batch; each thread handles 16 positions.
// Trivial work (8 blocks total) ahead of the bandwidth-bound gather.
// ---------------------------------------------------------------------------
__global__ void __launch_bounds__(256)
dechunk_cumsum_kernel(const unsigned char* __restrict__ mask,
                      int* __restrict__ idx) {
    const int b = blockIdx.x;
    const unsigned char* m = mask + (size_t)b * S_;
    int* out = idx + (size_t)b * S_;

    __shared__ int sums[256];
    const int t = threadIdx.x;
    const int per = S_ / 256;            // 16
    const int base = t * per;

    int local[per];
    int acc = 0;
#pragma unroll
    for (int i = 0; i < per; ++i) {
        acc += (m[base + i] != 0) ? 1 : 0;
        local[i] = acc;
    }
    sums[t] = acc;
    __syncthreads();

    // Serial exclusive prefix over 256 thread totals (trivial).
    if (t == 0) {
        int run = 0;
#pragma unroll 8
        for (int i = 0; i < 256; ++i) { int v = sums[i]; sums[i] = run; run += v; }
    }
    __syncthreads();

    const int off = sums[t];
#pragma unroll
    for (int i = 0; i < per; ++i) {
        out[base + i] = off + local[i] - 1;   // cum-1; -1 => pre-first-boundary
    }
}

// ---------------------------------------------------------------------------
// Host entry. Inputs (setup_inputs order):
//   d_in[0] compressed_states : float32 [B,S,D]
//   d_in[1] boundary_prob     : float32 [B,S,2]   (mathematically cancels; unused)
//   d_in[2] boundary_mask     : bool    [B,S]     (1 byte per element)
// Output: float32 [B,S,D].  Workspace: B*S ints = 128 KB of d_ws.
// ---------------------------------------------------------------------------
extern "C" void kernel_launch(void* const* d_in, const int* in_sizes, int n_in,
                              void* d_out, int out_size, void* d_ws, size_t ws_size,
                              hipStream_t stream) {
    const float* comp = (const float*)d_in[0];
    (void)d_in[1];
    const unsigned char* mask = (const unsigned char*)d_in[2];
    float* y = (float*)d_out;
    int* idx = (int*)d_ws;
    (void)in_sizes; (void)n_in; (void)out_size; (void)ws_size;

    dechunk_cumsum_kernel<<<B_, 256, 0, stream>>>(mask, idx);

    const int rows = B_ * S_;                       // 32768
    const int blocks = rows / WAVES_PER_BLOCK;      // 4096
    dechunk_gather_kernel<<<blocks, 32 * WAVES_PER_BLOCK, LDS_BYTES, stream>>>(comp, idx, y);
}